// DecNP_23407571763404
// MI455X (gfx1250) — compile-verified
//
#include <hip/hip_runtime.h>
#include <hip/hip_bf16.h>
#include <cstddef>

typedef __attribute__((ext_vector_type(2))) float v2f;
typedef __attribute__((ext_vector_type(8))) float v8f;

#define FLT_BIG 3.402823466e38f

// Branchless sorted insert of (d,s) into ascending triple (bd0<=bd1<=bd2).
__device__ __forceinline__ void insert3(float d, int s,
                                        float& bd0, float& bd1, float& bd2,
                                        int& bi0, int& bi1, int& bi2)
{
    const bool lt0 = d < bd0;
    const bool lt1 = d < bd1;
    const bool lt2 = d < bd2;
    const float nb2 = lt1 ? bd1 : (lt2 ? d : bd2);
    const int   ni2 = lt1 ? bi1 : (lt2 ? s : bi2);
    const float nb1 = lt0 ? bd0 : (lt1 ? d : bd1);
    const int   ni1 = lt0 ? bi0 : (lt1 ? s : bi1);
    bd0 = lt0 ? d : bd0;
    bi0 = lt0 ? s : bi0;
    bd1 = nb1; bi1 = ni1;
    bd2 = nb2; bi2 = ni2;
}

// ---------------------------------------------------------------------------
// Kernel 1: 3-NN (squared distance) via V_WMMA_F32_16X16X4_F32.
//
// A tile (16x4, M=coarse s, K=dim): lanes 0-15 hold row s=l with K0/K1 in the
// two VGPRs; lanes 16-31 hold row s=l with K2/K3.  LDS holds per coarse point
// the prepped row (-2x, -2y, -2z, |xc|^2).
// B tile (4x16, N=fine n): lanes 0-15 hold col n=l K0/K1; lanes 16-31 K2/K3.
// Fine column = (x, y, z, 1).
// => D[s][n] = -2*(xc.xf) + |xc|^2 ; add per-lane |xf|^2 for full d^2.
// D layout: lanes 0-15: VGPR r -> s=t*16+r, n=l; lanes 16-31: s=t*16+8+r.
// Each lane keeps a branchless top-3; halves merged with __shfl_xor(.,16).
// blockDim = 128 (4 waves = 64 fine points). grid = (N/64, B), exact.
// ---------------------------------------------------------------------------
__global__ __launch_bounds__(128) void knn3_wmma_kernel(
    const float* __restrict__ xyz_fine,    // [B, N, 3]
    const float* __restrict__ xyz_coarse,  // [B, S, 3]
    int N, int S,
    int*   __restrict__ out_idx,           // [B, N, 3]
    float* __restrict__ out_w)             // [B, N, 3]
{
    __shared__ float ls[1024 * 4];         // prepped coarse rows, S <= 1024

    const int b   = blockIdx.y;
    const int tid = threadIdx.x;

    for (int s = tid; s < S; s += 128) {
        const float* p = xyz_coarse + ((size_t)b * S + s) * 3;
        float x = p[0], y = p[1], z = p[2];
        ls[s * 4 + 0] = -2.0f * x;
        ls[s * 4 + 1] = -2.0f * y;
        ls[s * 4 + 2] = -2.0f * z;
        ls[s * 4 + 3] = x * x + y * y + z * z;
    }
    __syncthreads();

    const int wave = tid >> 5;
    const int lane = tid & 31;
    const int half = lane >> 4;
    const int l    = lane & 15;
    const int n    = blockIdx.x * 64 + wave * 16 + l;

    const float* pf = xyz_fine + ((size_t)b * N + n) * 3;
    const float fx = pf[0], fy = pf[1], fz = pf[2];
    const float fnrm = fx * fx + fy * fy + fz * fz;

    v2f Bf;
    Bf.x = half ? fz   : fx;
    Bf.y = half ? 1.0f : fy;

    float bd0 = FLT_BIG, bd1 = FLT_BIG, bd2 = FLT_BIG;
    int   bi0 = 0,       bi1 = 0,       bi2 = 0;

    const v2f* __restrict__ lsv = (const v2f*)ls;  // 2 x v2f per coarse row
    const int ntiles = S >> 4;

    // Software pipeline: fetch next tile's A fragment before scoring current.
    v2f Af = lsv[(size_t)l * 2 + half];
    for (int t = 0; t < ntiles; ++t) {
        v8f acc = {};
        // 8 args: (neg_a, A, neg_b, B, c_mod, C, reuse_a, reuse_b)
        acc = __builtin_amdgcn_wmma_f32_16x16x4_f32(
            false, Af, false, Bf, (short)0, acc, false, false);

        v2f Afn = Af;
        if (t + 1 < ntiles)                       // wave-uniform branch
            Afn = lsv[(size_t)((t + 1) * 16 + l) * 2 + half];

        const int sbase = t * 16 + half * 8;
#pragma unroll
        for (int r = 0; r < 8; ++r)
            insert3(acc[r] + fnrm, sbase + r, bd0, bd1, bd2, bi0, bi1, bi2);

        Af = Afn;
    }

    // Merge the two sorted triples held by lanes l and l+16 (branchless).
    float c0 = __shfl_xor(bd0, 16, 32);
    float c1 = __shfl_xor(bd1, 16, 32);
    float c2 = __shfl_xor(bd2, 16, 32);
    int   q0 = __shfl_xor(bi0, 16, 32);
    int   q1 = __shfl_xor(bi1, 16, 32);
    int   q2 = __shfl_xor(bi2, 16, 32);

    bool t0 = bd0 <= c0;
    const float m0 = t0 ? bd0 : c0;
    const int   j0 = t0 ? bi0 : q0;
    float a0 = t0 ? bd1 : bd0;  int ai0 = t0 ? bi1 : bi0;
    float a1 = t0 ? bd2 : bd1;  int ai1 = t0 ? bi2 : bi1;
    float e0 = t0 ? c0  : c1;   int ei0 = t0 ? q0  : q1;
    float e1 = t0 ? c1  : c2;   int ei1 = t0 ? q1  : q2;

    bool t1 = a0 <= e0;
    const float m1 = t1 ? a0 : e0;
    const int   j1 = t1 ? ai0 : ei0;
    float a0b = t1 ? a1 : a0;   int ai0b = t1 ? ai1 : ai0;
    float e0b = t1 ? e0 : e1;   int ei0b = t1 ? ei0 : ei1;

    bool t2 = a0b <= e0b;
    const float m2 = t2 ? a0b : e0b;
    const int   j2 = t2 ? ai0b : ei0b;

    if (half == 0) {
        float w0 = 1.0f / (m0 + 1e-8f);
        float w1 = 1.0f / (m1 + 1e-8f);
        float w2 = 1.0f / (m2 + 1e-8f);
        float inv = 1.0f / (w0 + w1 + w2);
        size_t base = ((size_t)b * N + n) * 3;
        out_idx[base + 0] = j0;
        out_idx[base + 1] = j1;
        out_idx[base + 2] = j2;
        out_w[base + 0] = w0 * inv;
        out_w[base + 1] = w1 * inv;
        out_w[base + 2] = w2 * inv;
    }
}

// ---------------------------------------------------------------------------
// Kernel 2: out[b, 0:C1, n] = skip[b, :, n];
//           out[b, C1+c2, n] = sum_k w[b,n,k] * prev[b, c2, idx[b,n,k]]
// Block = 256 threads = 16 n-quads x 16 channel groups; each thread owns
// 4 consecutive n -> float4 (B128) stores, float4 skip loads.
// idx/w staged in LDS.  grid = (N/64, B).  All C totals / C1 divide by 16,
// so the c < C1 branch is uniform per iteration.
// ---------------------------------------------------------------------------
__global__ __launch_bounds__(256) void interp_concat_kernel(
    const float* __restrict__ skip, int C1,   // [B, C1, N]
    const float* __restrict__ prev, int C2,   // [B, C2, S]
    int S,
    const int*   __restrict__ idx,            // [B, N, 3]
    const float* __restrict__ w,              // [B, N, 3]
    float* __restrict__ out,                  // [B, C1+C2, N]
    int N)
{
    __shared__ int   sidx[64 * 3];
    __shared__ float sw[64 * 3];

    const int b   = blockIdx.y;
    const int n0  = blockIdx.x * 64;
    const int tid = threadIdx.x;

    for (int i = tid; i < 192; i += 256) {
        size_t base = ((size_t)b * N + n0) * 3 + i;
        sidx[i] = idx[base];
        sw[i]   = w[base];
    }
    __syncthreads();

    const int nq   = (tid & 15) * 4;   // local n-quad base: 0..60
    const int cofs = tid >> 4;         // 0..15
    const int n    = n0 + nq;
    const int Ct   = C1 + C2;

    int   ii[4][3];
    float ww[4][3];
#pragma unroll
    for (int j = 0; j < 4; ++j) {
#pragma unroll
        for (int k = 0; k < 3; ++k) {
            ii[j][k] = sidx[(nq + j) * 3 + k];
            ww[j][k] = sw[(nq + j) * 3 + k];
        }
    }

    for (int c = cofs; c < Ct; c += 16) {
        float4 v;
        if (c < C1) {
            v = *(const float4*)(skip + ((size_t)b * C1 + c) * N + n);
        } else {
            const float* __restrict__ row = prev + ((size_t)b * C2 + (c - C1)) * S;
            float res[4];
#pragma unroll
            for (int j = 0; j < 4; ++j)
                res[j] = ww[j][0] * row[ii[j][0]]
                       + ww[j][1] * row[ii[j][1]]
                       + ww[j][2] * row[ii[j][2]];
            v = make_float4(res[0], res[1], res[2], res[3]);
        }
        *(float4*)(out + ((size_t)b * Ct + c) * N + n) = v;
    }
}

// ---------------------------------------------------------------------------
// Host side
// ---------------------------------------------------------------------------
static inline size_t ws_align(size_t x) { return (x + 255) & ~(size_t)255; }

extern "C" void kernel_launch(void* const* d_in, const int* in_sizes, int n_in,
                              void* d_out, int out_size, void* d_ws, size_t ws_size,
                              hipStream_t stream) {
    (void)in_sizes; (void)n_in; (void)out_size; (void)ws_size;

    // setup_inputs order: xyz0, f0, xyz1, f1, xyz2, f2, xyz3, f3, xyz4, f4
    const float* xyz0 = (const float*)d_in[0];
    const float* f0   = (const float*)d_in[1];
    const float* xyz1 = (const float*)d_in[2];
    const float* f1   = (const float*)d_in[3];
    const float* xyz2 = (const float*)d_in[4];
    const float* f2   = (const float*)d_in[5];
    const float* xyz3 = (const float*)d_in[6];
    const float* f3   = (const float*)d_in[7];
    const float* xyz4 = (const float*)d_in[8];
    const float* f4   = (const float*)d_in[9];

    const int B = 16;

    char* ws = (char*)d_ws;
    size_t off = 0;
    int*   idxbuf = (int*)(ws + off);   off += ws_align((size_t)B * 4096 * 3 * sizeof(int));
    float* wbuf   = (float*)(ws + off); off += ws_align((size_t)B * 4096 * 3 * sizeof(float));
    float* out0   = (float*)(ws + off); off += ws_align((size_t)B * 1536 * 64   * sizeof(float));
    float* out1   = (float*)(ws + off); off += ws_align((size_t)B * 1792 * 256  * sizeof(float));
    float* out2   = (float*)(ws + off); off += ws_align((size_t)B * 1920 * 1024 * sizeof(float));

    struct Stage {
        const float* xf; const float* xc; int N; int S;
        const float* skip; int C1; const float* prev; int C2; float* out;
    } st[4] = {
        { xyz3, xyz4,   64,   16, f3, 512, f4,   1024, out0 },
        { xyz2, xyz3,  256,   64, f2, 256, out0, 1536, out1 },
        { xyz1, xyz2, 1024,  256, f1, 128, out1, 1792, out2 },
        { xyz0, xyz1, 4096, 1024, f0,  64, out2, 1920, (float*)d_out },
    };

    for (int i = 0; i < 4; ++i) {
        dim3 grid(st[i].N / 64, B);
        knn3_wmma_kernel<<<grid, 128, 0, stream>>>(
            st[i].xf, st[i].xc, st[i].N, st[i].S, idxbuf, wbuf);
        interp_concat_kernel<<<grid, 256, 0, stream>>>(
            st[i].skip, st[i].C1, st[i].prev, st[i].C2, st[i].S,
            idxbuf, wbuf, st[i].out, st[i].N);
    }
}